// GeometricProductLayer_87333864997057
// MI455X (gfx1250) — compile-verified
//
#include <hip/hip_runtime.h>

typedef __attribute__((ext_vector_type(2))) float v2f;
typedef __attribute__((ext_vector_type(4))) float v4f;
typedef __attribute__((ext_vector_type(8))) float v8f;

// ---------------------------------------------------------------------------
// Compile-time enumeration of the 192 nonzero Cayley pairs (i,j).
// Basis masks follow the reference _USER_BASIS order; a product vanishes iff
// both blades contain the null e0 (mask bit 0).
// ---------------------------------------------------------------------------
struct Pairs { unsigned char i[192]; unsigned char j[192]; };

constexpr Pairs make_pairs() {
  Pairs p{};
  const int m[16] = {0, 2, 4, 8, 1, 6, 10, 12, 3, 5, 9, 14, 7, 11, 13, 15};
  int n = 0;
  for (int i = 0; i < 16; ++i)
    for (int j = 0; j < 16; ++j)
      if (((m[i] & m[j]) & 1) == 0) {
        p.i[n] = (unsigned char)i;
        p.j[n] = (unsigned char)j;
        ++n;
      }
  return p;
}
constexpr Pairs PP = make_pairs();   // exactly 192 pairs -> 48 K=4 groups

// ---------------------------------------------------------------------------
// Per tile of 16 multivectors:
//   C_tile[16n x 16k] = P[16n x 192] * W[192 x 16k]
// with P[n,p] = A[n,i_p]*B[n,j_p] and W[p,k] = cayley[k, i_p, j_p] (+-1 / 0).
// 48x V_WMMA_F32_16X16X4_F32, f32 accumulate (exact vs reference).
// ---------------------------------------------------------------------------
__global__ __launch_bounds__(256) void gp_wmma_kernel(
    const float* __restrict__ A, const float* __restrict__ B,
    const float* __restrict__ T,  // cayley [16,16,16], k-major
    float* __restrict__ C, int nTiles) {
  const int lane = threadIdx.x & 31;
  const bool hi  = lane >= 16;     // upper half-wave: K slots 2,3 / rows M+8
  const int  kk  = lane & 15;      // row (for A/B data) and k column (for W/D)

  // ---- Load the 48 constant B-operands (W slices) straight from the table.
  // B operand layout (4x16 f32): VGPR0 = {K0 | K2}, VGPR1 = {K1 | K3},
  // lanes 0-15 / 16-31; lane kk needs W[p,kk] = cayley[kk*256 + i_p*16 + j_p].
  v2f bop[48];
#pragma unroll
  for (int g = 0; g < 48; ++g) {
    const int p0 = 4 * g, p1 = 4 * g + 1, p2 = 4 * g + 2, p3 = 4 * g + 3;
    const int ox = kk * 256 + (hi ? (PP.i[p2] * 16 + PP.j[p2])
                                  : (PP.i[p0] * 16 + PP.j[p0]));
    const int oy = kk * 256 + (hi ? (PP.i[p3] * 16 + PP.j[p3])
                                  : (PP.i[p1] * 16 + PP.j[p1]));
    bop[g].x = T[ox];
    bop[g].y = T[oy];
  }

  const int waveId = (int)((blockIdx.x * blockDim.x + threadIdx.x) >> 5);
  const int nWaves = (int)((gridDim.x * blockDim.x) >> 5);

  for (int t = waveId; t < nTiles; t += nWaves) {
    const long base = (long)t * 256;          // 16 rows x 16 comps
    const float* ar = A + base + kk * 16;     // this lane's multivector row
    const float* br = B + base + kk * 16;     // (upper half duplicates rows,
                                              //  as the A-operand layout needs)
    float a[16], b[16];
#pragma unroll
    for (int q = 0; q < 4; ++q) {
      v4f av = *(const v4f*)(ar + 4 * q);
      v4f bv = *(const v4f*)(br + 4 * q);
      a[4 * q + 0] = av.x; a[4 * q + 1] = av.y;
      a[4 * q + 2] = av.z; a[4 * q + 3] = av.w;
      b[4 * q + 0] = bv.x; b[4 * q + 1] = bv.y;
      b[4 * q + 2] = bv.z; b[4 * q + 3] = bv.w;
    }

    v8f acc = {0.f, 0.f, 0.f, 0.f, 0.f, 0.f, 0.f, 0.f};
#pragma unroll
    for (int g = 0; g < 48; ++g) {
      // A operand (16x4): VGPR0 = {K0 | K2}, VGPR1 = {K1 | K3}
      v2f aop;
      aop.x = hi ? a[PP.i[4 * g + 2]] * b[PP.j[4 * g + 2]]
                 : a[PP.i[4 * g + 0]] * b[PP.j[4 * g + 0]];
      aop.y = hi ? a[PP.i[4 * g + 3]] * b[PP.j[4 * g + 3]]
                 : a[PP.i[4 * g + 1]] * b[PP.j[4 * g + 1]];
      acc = __builtin_amdgcn_wmma_f32_16x16x4_f32(
          /*neg_a=*/false, aop, /*neg_b=*/false, bop[g],
          /*c_mod=*/(short)0, acc, /*reuse_a=*/false, /*reuse_b=*/false);
    }

    // D layout: lane L, VGPR v -> C[row = v + 8*(L>=16), col = L&15]
    float* cr = C + base + kk;
    const int rowoff = hi ? 8 : 0;
#pragma unroll
    for (int v = 0; v < 8; ++v) {
      cr[(long)(v + rowoff) * 16] = acc[v];
    }
  }
}

extern "C" void kernel_launch(void* const* d_in, const int* in_sizes, int n_in,
                              void* d_out, int out_size, void* d_ws, size_t ws_size,
                              hipStream_t stream) {
  (void)in_sizes; (void)n_in; (void)d_ws; (void)ws_size;
  const float* mv_a   = (const float*)d_in[0];
  const float* mv_b   = (const float*)d_in[1];
  const float* cayley = (const float*)d_in[2];
  float* out = (float*)d_out;

  const int nTiles = out_size / 256;          // 16 multivectors per tile
  // Persistent waves: ~8 tiles per wave so the 96 table-gather loads amortize.
  int waves  = (nTiles + 7) / 8;
  int blocks = (waves * 32 + 255) / 256;
  if (blocks < 1) blocks = 1;
  if (blocks > 16384) blocks = 16384;

  gp_wmma_kernel<<<blocks, 256, 0, stream>>>(mv_a, mv_b, cayley, out, nTiles);
}